// GCNLayer_22892175687762
// MI455X (gfx1250) — compile-verified
//
#include <hip/hip_runtime.h>
#include <hip/hip_bf16.h>

typedef __attribute__((ext_vector_type(2))) float v2f;
typedef __attribute__((ext_vector_type(8))) float v8f;

#define D_FEAT 128

// ---------------------------------------------------------------------------
// Kernel 1: zero the fp32 accumulator workspace h[N, 128]
// ---------------------------------------------------------------------------
__global__ void gcn_zero_f4(float4* __restrict__ p, size_t n4) {
    size_t i = (size_t)blockIdx.x * blockDim.x + threadIdx.x;
    size_t stride = (size_t)gridDim.x * blockDim.x;
    for (; i < n4; i += stride) {
        p[i] = make_float4(0.f, 0.f, 0.f, 0.f);
    }
}

// ---------------------------------------------------------------------------
// Kernel 2: edge scatter  h[dst_e] += feature[src_e] * w_e
// One wave32 per edge; lane l handles features [4l, 4l+4).
// feature (51.2MB) and h (51.2MB) both fit in the 192MB L2, so the 204.8M
// f32 atomics resolve in L2; gather is one coalesced 512B read per wave.
// ---------------------------------------------------------------------------
__global__ void gcn_edge_scatter(const float* __restrict__ feat,
                                 const int* __restrict__ src,
                                 const int* __restrict__ dst,
                                 const float* __restrict__ ew,
                                 float* __restrict__ h,
                                 int n_edges) {
    long tid = (long)blockIdx.x * blockDim.x + threadIdx.x;
    int e = (int)(tid >> 5);
    if (e >= n_edges) return;
    int lane = (int)(tid & 31);

    int s = src[e];
    int d = dst[e];
    float w = ew[e];

    const float4 f = *(const float4*)(feat + (size_t)s * D_FEAT + lane * 4);
    float* hp = h + (size_t)d * D_FEAT + lane * 4;
    atomicAdd(hp + 0, f.x * w);
    atomicAdd(hp + 1, f.y * w);
    atomicAdd(hp + 2, f.z * w);
    atomicAdd(hp + 3, f.w * w);
}

// ---------------------------------------------------------------------------
// Kernel 3: y = h @ W^T + b via V_WMMA_F32_16X16X4_F32 (fp32, wave32).
//
// Block = 256 threads = 8 waves. Block b -> rows [32b, 32b+32) (two 16-row
// tiles per wave); wave w -> cols [16w, 16w+16)  (D_OUT = 128 = 8x16).
// Each B float2 load is reused by two WMMAs (two M tiles).
//
// A (16x4 f32) lane layout: lanes 0-15 hold row M=lane, V0=K0,V1=K1;
//                           lanes 16-31 hold row M=lane-16, V0=K2,V1=K3.
// B (4x16) transposed analog; B[k][n] = W[n][k] so each lane reads float2s
// from row (n0 + lane&15) of W — same access shape as A.
// C/D: VGPR r, lanes 0-15 -> (m+r, n0+l); lanes 16-31 -> (m+r+8, n0+l).
// ---------------------------------------------------------------------------
__global__ void gcn_gemm_bias_wmma(const float* __restrict__ h,
                                   const float* __restrict__ W,
                                   const float* __restrict__ bias,
                                   float* __restrict__ y,
                                   int n_rows) {
    const int lane = threadIdx.x & 31;
    const int wave = threadIdx.x >> 5;          // 0..7 -> N tile
    const int m0 = blockIdx.x * 32;             // two 16-row M tiles
    const int n0 = wave * 16;
    const bool hi = lane >= 16;
    const int l = lane & 15;
    const int kbase = hi ? 2 : 0;

    // Clamp A rows so loads stay in bounds for ragged sizes while EXEC stays
    // all-ones (WMMA requirement). Stores handle raggedness separately.
    int arow0 = m0 + l;
    int arow1 = m0 + 16 + l;
    if (arow0 >= n_rows) arow0 = n_rows - 1;
    if (arow1 >= n_rows) arow1 = n_rows - 1;

    const float* ap0 = h + (size_t)arow0 * D_FEAT + kbase;
    const float* ap1 = h + (size_t)arow1 * D_FEAT + kbase;
    const float* bp  = W + (size_t)(n0 + l) * D_FEAT + kbase;

    v8f c0 = {};
    v8f c1 = {};
#pragma unroll 8
    for (int k0 = 0; k0 < D_FEAT; k0 += 4) {
        v2f b  = *(const v2f*)(bp  + k0);   // 8B-aligned (kbase even)
        v2f a0 = *(const v2f*)(ap0 + k0);
        v2f a1 = *(const v2f*)(ap1 + k0);
        // (neg_a, A, neg_b, B, c_mod, C, reuse_a, reuse_b)
        c0 = __builtin_amdgcn_wmma_f32_16x16x4_f32(
                 false, a0, false, b, (short)0, c0, false, false);
        c1 = __builtin_amdgcn_wmma_f32_16x16x4_f32(
                 false, a1, false, b, (short)0, c1, false, false);
    }

    const float bcol = bias[n0 + l];            // output column for this lane
    const int r0 = m0 + (hi ? 8 : 0);
    const int r1 = m0 + 16 + (hi ? 8 : 0);
    float* out0 = y + (size_t)r0 * D_FEAT + n0 + l;
    float* out1 = y + (size_t)r1 * D_FEAT + n0 + l;

    if (m0 + 32 <= n_rows) {
        // Fast path: full tile, no per-row predication (the hot path; for
        // N=100000 every block takes it since 100000 % 32 == 0).
#pragma unroll
        for (int r = 0; r < 8; ++r) {
            out0[(size_t)r * D_FEAT] = c0[r] + bcol;
            out1[(size_t)r * D_FEAT] = c1[r] + bcol;
        }
    } else {
#pragma unroll
        for (int r = 0; r < 8; ++r) {
            if (r0 + r < n_rows) out0[(size_t)r * D_FEAT] = c0[r] + bcol;
            if (r1 + r < n_rows) out1[(size_t)r * D_FEAT] = c1[r] + bcol;
        }
    }
}

// ---------------------------------------------------------------------------
extern "C" void kernel_launch(void* const* d_in, const int* in_sizes, int n_in,
                              void* d_out, int out_size, void* d_ws, size_t ws_size,
                              hipStream_t stream) {
    const float* feat = (const float*)d_in[0];   // [N, 128]
    const int*   src  = (const int*)d_in[1];     // [E]
    const int*   dst  = (const int*)d_in[2];     // [E]
    const float* ew   = (const float*)d_in[3];   // [E]
    const float* W    = (const float*)d_in[4];   // [128, 128]
    const float* bias = (const float*)d_in[5];   // [128]
    float* y = (float*)d_out;                    // [N, 128]
    float* h = (float*)d_ws;                     // [N, 128] fp32 accumulator

    const int N = in_sizes[0] / D_FEAT;          // 100000
    const int E = in_sizes[1];                   // 1600000

    // 1) zero accumulator
    size_t n4 = (size_t)N * D_FEAT / 4;
    int zgrid = (int)((n4 + 255) / 256);
    if (zgrid > 4096) zgrid = 4096;
    gcn_zero_f4<<<zgrid, 256, 0, stream>>>((float4*)h, n4);

    // 2) edge scatter: one wave per edge
    long total = (long)E * 32;
    int sgrid = (int)((total + 255) / 256);
    gcn_edge_scatter<<<sgrid, 256, 0, stream>>>(feat, src, dst, ew, h, E);

    // 3) WMMA GEMM + bias: 32 rows per block, 3125 blocks
    int mtiles = (N + 31) / 32;
    gcn_gemm_bias_wmma<<<mtiles, 256, 0, stream>>>(h, W, bias, y, N);
}